// BoundaryLoss_88210038325619
// MI455X (gfx1250) — compile-verified
//
#include <hip/hip_runtime.h>
#include <hip/hip_bf16.h>

#ifndef __has_builtin
#define __has_builtin(x) 0
#endif
#ifndef __has_include
#define __has_include(x) 0
#endif

#define IMG_B 32
#define IMG_H 512
#define IMG_W 512
#define TH 32
#define TW 64
#define PTH (TH + 4)          // 36 padded rows
#define PTW (TW + 4)          // 68 padded cols
#define TILES_X (IMG_W / TW)  // 8
#define TILES_Y (IMG_H / TH)  // 16
#define TILES_PER_IMG (TILES_X * TILES_Y)  // 128
#define NBLOCKS (IMG_B * TILES_PER_IMG)    // 4096
#define NTHREADS 256
#define N_ELEMS 8388608.0     // 32*512*512

typedef unsigned int u32x4 __attribute__((ext_vector_type(4)));
typedef int i32x4 __attribute__((ext_vector_type(4)));
typedef int i32x8 __attribute__((ext_vector_type(8)));

// --- Tensor Data Mover availability / arity selection (per CDNA5_HIP.md) ---
#if defined(__HIP_DEVICE_COMPILE__)
#if __has_builtin(__builtin_amdgcn_tensor_load_to_lds)
#define HAVE_TDM 1
#endif
#endif
#ifndef HAVE_TDM
#define HAVE_TDM 0
#endif

#if HAVE_TDM
#if __has_include(<hip/amd_detail/amd_gfx1250_TDM.h>)
#define TDM_6ARG 1   // amdgpu-toolchain therock-10.0 headers -> 6-arg builtin
#endif

// Issue one TDM op: load `rows` rows of `cols` f32 elements from global
// (row pitch IMG_W elements) into LDS at byte offset lds_byte_addr with LDS
// row pitch PTW elements, using descriptor-iteration mode (one 1-row tile per
// iteration so the LDS pitch can be arbitrary).
__device__ __forceinline__ void tdm_load_rows(unsigned lds_byte_addr,
                                              unsigned long long gaddr,
                                              int cols, int rows) {
  // D# group 0: count=1 | lds_addr | global_addr[56:0] | type=2
  u32x4 g0;
  g0.x = 1u;
  g0.y = lds_byte_addr;
  g0.z = (unsigned)(gaddr & 0xFFFFFFFFull);
  g0.w = (unsigned)((gaddr >> 32) & 0x01FFFFFFull) | (2u << 30);

  // D# group 1
  i32x8 g1;
  g1.s0 = (2 << 16) | (1 << 19);               // data_size=4B, iterate_enable=1
  g1.s1 = (int)((cols & 0xFFFF) << 16);        // tensor_dim0[15:0] = cols
  g1.s2 = (int)(((cols >> 16) & 0xFFFF) | (1u << 16)); // dim0 hi | tensor_dim1 = 1
  g1.s3 = (int)((cols & 0xFFFF) << 16);        // tensor_dim1 hi=0 | tile_dim0 = cols
  g1.s4 = 1;                                   // tile_dim1 = 1, tile_dim2 = 0
  g1.s5 = IMG_W;                               // tensor_dim0_stride (elements)
  g1.s6 = 0;
  g1.s7 = 0;

  // D# group 2 (iterate mode): lds/global increments + iteration count
  i32x4 g2;
  g2.x = 0;
  g2.y = PTW;                                  // lds_addr_increment (elements)
  g2.z = IMG_W;                                // global_addr_increment (elements)
  g2.w = (int)((rows - 1) << 16);              // iterate_count (0 => 1x)

  i32x4 g3 = {0, 0, 0, 0};

#if defined(TDM_6ARG)
  i32x8 z8 = {0, 0, 0, 0, 0, 0, 0, 0};
  __builtin_amdgcn_tensor_load_to_lds(g0, g1, g2, g3, z8, 0);
#else
  __builtin_amdgcn_tensor_load_to_lds(g0, g1, g2, g3, 0);
#endif
}
#endif  // HAVE_TDM

struct __align__(16) Tile {
  float t[PTH][PTW];       // padded target tile (offset 0 in LDS)
  float cmin[TH][PTW];     // vertical 5-tap min
  float cmax[TH][PTW];     // vertical 5-tap max
  double red[NTHREADS / 32];
};

__device__ __forceinline__ float softplusf(float x) {
  // log1p(exp(x)) stable; equals -log(sigmoid(-x))
  float ax = fabsf(x);
  return fmaxf(x, 0.0f) + log1pf(__expf(-ax));
}

__global__ __launch_bounds__(NTHREADS)
void boundary_bce_partial(const float* __restrict__ pred,
                          const float* __restrict__ target,
                          double* __restrict__ partial) {
  __shared__ Tile sm;
  const int tid = threadIdx.x;
  const int bid = blockIdx.x;
  const int img = bid / TILES_PER_IMG;
  const int rem = bid - img * TILES_PER_IMG;
  const int tiy = rem / TILES_X;
  const int tix = rem - tiy * TILES_X;
  const int y0 = tiy * TH;
  const int x0 = tix * TW;

  // 1) zero padded tile => zero-padding semantics of the reference box-sum
  float* tflat = &sm.t[0][0];
  for (int i = tid; i < PTH * PTW; i += NTHREADS) tflat[i] = 0.0f;
  __syncthreads();

  // 2) clipped halo rectangle (global coords) -> LDS via TDM
  const int cys = max(y0 - 2, 0);
  const int cxs = max(x0 - 2, 0);
  const int cye = min(y0 + TH + 2, IMG_H);
  const int cxe = min(x0 + TW + 2, IMG_W);
  const int rows = cye - cys;
  const int cols = cxe - cxs;
  const int lr0 = cys - (y0 - 2);  // 0..2
  const int lc0 = cxs - (x0 - 2);  // 0..2
  const size_t gbase = ((size_t)img * IMG_H + cys) * IMG_W + cxs;

#if HAVE_TDM
  // Two tensor DMAs per block: wave 0 loads the top half of the rows,
  // wave 1 the bottom half (disjoint LDS ranges, independent TENSORcnt).
  {
    const int half = rows >> 1;
    if (tid < 32) {
      unsigned lds_addr = (unsigned)((lr0 * PTW + lc0) * 4);  // Tile.t at LDS 0
      tdm_load_rows(lds_addr,
                    (unsigned long long)(const void*)(target + gbase),
                    cols, half);
      __builtin_amdgcn_s_wait_tensorcnt(0);
    } else if (tid < 64) {
      unsigned lds_addr = (unsigned)(((lr0 + half) * PTW + lc0) * 4);
      tdm_load_rows(lds_addr,
                    (unsigned long long)(const void*)(target + gbase +
                                                      (size_t)half * IMG_W),
                    cols, rows - half);
      __builtin_amdgcn_s_wait_tensorcnt(0);
    }
  }
#else
  for (int i = tid; i < rows * cols; i += NTHREADS) {
    int rr = i / cols, cc = i - rr * cols;
    sm.t[lr0 + rr][lc0 + cc] = target[gbase + (size_t)rr * IMG_W + cc];
  }
#endif
  __syncthreads();

  // 3) vertical 5-tap min/max (binary target: box-sum tests == min/max tests)
  for (int i = tid; i < TH * PTW; i += NTHREADS) {
    int r = i / PTW, c = i - r * PTW;
    float a0 = sm.t[r + 0][c], a1 = sm.t[r + 1][c], a2 = sm.t[r + 2][c];
    float a3 = sm.t[r + 3][c], a4 = sm.t[r + 4][c];
    sm.cmin[r][c] = fminf(fminf(fminf(a0, a1), fminf(a2, a3)), a4);
    sm.cmax[r][c] = fmaxf(fmaxf(fmaxf(a0, a1), fmaxf(a2, a3)), a4);
  }
  __syncthreads();

  // 4) horizontal 5-tap + fused BCE + weight, accumulate in double
  double acc = 0.0;
  const int c = tid % TW;          // 0..63 (coalesced pred loads)
  const int rb = tid / TW;         // 0..3
  const size_t prow0 = ((size_t)img * IMG_H + y0) * IMG_W + x0 + c;
#pragma unroll
  for (int k = 0; k < TH / (NTHREADS / TW); ++k) {  // 8 pixels/thread
    const int r = k * (NTHREADS / TW) + rb;
    float mn = fminf(fminf(fminf(sm.cmin[r][c], sm.cmin[r][c + 1]),
                           fminf(sm.cmin[r][c + 2], sm.cmin[r][c + 3])),
                     sm.cmin[r][c + 4]);
    float mx = fmaxf(fmaxf(fmaxf(sm.cmax[r][c], sm.cmax[r][c + 1]),
                           fmaxf(sm.cmax[r][c + 2], sm.cmax[r][c + 3])),
                     sm.cmax[r][c + 4]);
    // dilated=(s>0) <=> mx==1 ; eroded=(s==25) <=> mn==1 (OOB counts as 0)
    float w = (mx > 0.5f && mn < 0.5f) ? 5.0f : 1.0f;  // 4*boundary + 1
    float tt = sm.t[r + 2][c + 2];
    float p = pred[prow0 + (size_t)r * IMG_W];
    float bce = tt * fminf(softplusf(-p), 100.0f)
              + (1.0f - tt) * fminf(softplusf(p), 100.0f);
    acc += (double)(bce * w);
  }

  // 5) block reduction (wave32 shuffle + LDS)
  for (int off = 16; off > 0; off >>= 1) acc += __shfl_down(acc, off, 32);
  const int wave = tid >> 5, lane = tid & 31;
  if (lane == 0) sm.red[wave] = acc;
  __syncthreads();
  if (tid == 0) {
    double s = 0.0;
    for (int wv = 0; wv < NTHREADS / 32; ++wv) s += sm.red[wv];
    partial[bid] = s;
  }
}

__global__ __launch_bounds__(NTHREADS)
void reduce_mean(const double* __restrict__ partial, float* __restrict__ out,
                 int n) {
  __shared__ double red[NTHREADS / 32];
  double s = 0.0;
  for (int i = threadIdx.x; i < n; i += NTHREADS) s += partial[i];
  for (int off = 16; off > 0; off >>= 1) s += __shfl_down(s, off, 32);
  const int wave = threadIdx.x >> 5, lane = threadIdx.x & 31;
  if (lane == 0) red[wave] = s;
  __syncthreads();
  if (threadIdx.x == 0) {
    double tot = 0.0;
    for (int wv = 0; wv < NTHREADS / 32; ++wv) tot += red[wv];
    out[0] = (float)(tot / N_ELEMS);
  }
}

extern "C" void kernel_launch(void* const* d_in, const int* in_sizes, int n_in,
                              void* d_out, int out_size, void* d_ws,
                              size_t ws_size, hipStream_t stream) {
  (void)in_sizes; (void)n_in; (void)out_size; (void)ws_size;
  const float* pred = (const float*)d_in[0];
  const float* target = (const float*)d_in[1];
  float* out = (float*)d_out;
  double* partial = (double*)d_ws;  // NBLOCKS doubles = 32 KB scratch

  boundary_bce_partial<<<NBLOCKS, NTHREADS, 0, stream>>>(pred, target, partial);
  reduce_mean<<<1, NTHREADS, 0, stream>>>(partial, out, NBLOCKS);
}